// EGNN_67800353734913
// MI455X (gfx1250) — compile-verified
//
#include <hip/hip_runtime.h>
#include <math.h>

// ---------------- problem constants ----------------
#define Bk   4
#define Nk   160
#define Fk   16
#define Hk   128
#define Lk   4
#define TI   16           // i-tile
#define TJ   8            // j-tile
#define EDG  (TI*TJ)      // 128 edges per block
#define NTI  (Nk/TI)      // 10
#define NTJ  (Nk/TJ)      // 20
#define SMN  132          // padded LDS row stride (floats); 132%64=4 -> conflict-free
#define CRANGE 3.75f      // 15.0 / L

typedef __attribute__((ext_vector_type(2))) float v2f;
typedef __attribute__((ext_vector_type(8))) float v8f;

// fast sigmoid / SiLU: v_exp_f32 + v_rcp_f32 (avoids the IEEE div expansion)
__device__ __forceinline__ float fast_sigmoid(float s) {
    return __builtin_amdgcn_rcpf(1.0f + __expf(-s));
}
__device__ __forceinline__ float fast_silu(float s) {
    return s * fast_sigmoid(s);
}

// 128x128 = (128 edge-rows) x (128 cols) GEMM + bias + SiLU, via f32 WMMA 16x16x4.
// src/w/dst are LDS arrays with row stride SMN. All 256 threads call uniformly
// (EXEC all-ones requirement for WMMA).
__device__ __forceinline__ void tile_mm_silu(const float* src, const float* w,
                                             const float* __restrict__ bias,
                                             float* dst, int tid)
{
    const int lane    = tid & 31;
    const int wave    = tid >> 5;              // 8 waves -> 8 row tiles
    const int r       = lane & 15;
    const int kh      = (lane < 16) ? 0 : 2;   // K half per lane group
    const int rowOff  = (lane < 16) ? 0 : 8;   // C/D row split
    const int rowBase = wave * 16;
    const float* arow = src + (size_t)(rowBase + r) * SMN;
    #pragma unroll
    for (int ct = 0; ct < 8; ++ct) {
        const int cb = ct * 16;
        const float bv = bias[cb + r];
        v8f acc;
        #pragma unroll
        for (int v = 0; v < 8; ++v) acc[v] = bv;
        #pragma unroll 4
        for (int k = 0; k < Hk; k += 4) {
            v2f a, b;
            a.x = arow[k + kh];
            a.y = arow[k + kh + 1];
            b.x = w[(size_t)(k + kh) * SMN + cb + r];
            b.y = w[(size_t)(k + kh + 1) * SMN + cb + r];
            acc = __builtin_amdgcn_wmma_f32_16x16x4_f32(
                      false, a, false, b, (short)0, acc, false, false);
        }
        #pragma unroll
        for (int v = 0; v < 8; ++v) {
            float s = acc[v];
            dst[(size_t)(rowBase + rowOff + v) * SMN + cb + r] = fast_silu(s);
        }
    }
}

// ---------------- small prep kernels ----------------
__global__ __launch_bounds__(256) void embed_kernel(
    const float* __restrict__ h, const float* __restrict__ t,
    const float* __restrict__ nmask, const float* __restrict__ W_in,
    const float* __restrict__ b_in, float* __restrict__ hcur)
{
    int gid = blockIdx.x * 256 + threadIdx.x;      // < B*N*H = 81920
    int n = gid >> 7, f = gid & 127;
    int b = n / Nk;
    float s = b_in[f] + t[b] * W_in[(size_t)Fk * Hk + f];
    const float* hr = h + (size_t)n * Fk;
    #pragma unroll
    for (int k = 0; k < Fk; ++k) s += hr[k] * W_in[(size_t)k * Hk + f];
    hcur[gid] = s * nmask[n];
}

__global__ __launch_bounds__(256) void dist_kernel(
    const float* __restrict__ x, float* __restrict__ d2, float* __restrict__ xcur)
{
    int gid = blockIdx.x * 256 + threadIdx.x;      // < B*N*N = 102400
    if (gid < Bk * Nk * 3) xcur[gid] = x[gid];
    int j = gid % Nk; int r = gid / Nk; int i = r % Nk; int b = r / Nk;
    const float* xi = x + ((size_t)b * Nk + i) * 3;
    const float* xj = x + ((size_t)b * Nk + j) * 3;
    float d0 = xi[0]-xj[0], d1 = xi[1]-xj[1], dd = xi[2]-xj[2];
    d2[gid] = d0*d0 + d1*d1 + dd*dd;
}

// aT = hcur @ We1[0:128] + be1   (per target i)
// bT = hcur @ We1[128:256]       (per source j)
__global__ __launch_bounds__(256) void proj_kernel(
    const float* __restrict__ hcur, const float* __restrict__ We1,
    const float* __restrict__ be1, float* __restrict__ projA, float* __restrict__ projB)
{
    int gid = blockIdx.x * 256 + threadIdx.x;      // < 2*B*N*H
    const int half = Bk * Nk * Hk;
    bool isB = gid >= half;
    int g = isB ? gid - half : gid;
    int n = g >> 7, f = g & 127;
    const float* hr = hcur + (size_t)n * Hk;
    const float* wcol = We1 + (isB ? (size_t)Hk * Hk : 0) + f;
    float s = isB ? 0.0f : be1[f];
    for (int k = 0; k < Hk; ++k) s += hr[k] * wcol[(size_t)k * Hk];
    (isB ? projB : projA)[g] = s;
}

// ---------------- fused edge kernel ----------------
__global__ __launch_bounds__(256) void edge_kernel(
    const float* __restrict__ projA, const float* __restrict__ projB,
    const float* __restrict__ d2,    const float* __restrict__ xcur,
    const float* __restrict__ emask, const float* __restrict__ wd,
    const float* __restrict__ We2,   const float* __restrict__ be2,
    const float* __restrict__ Wa,    const float* __restrict__ ba,
    const float* __restrict__ Wc1,   const float* __restrict__ bc1,
    const float* __restrict__ Wc2,
    float* __restrict__ aggp, float* __restrict__ xp)
{
    __shared__ float sM1[EDG][SMN];   // pre/m1, later c
    __shared__ float sM2[EDG][SMN];   // m2s -> gated m
    __shared__ float sW [Hk][SMN];    // We2, later Wc1
    __shared__ float sVec[Hk];        // wd -> Wa -> Wc2
    __shared__ float sD [EDG];
    __shared__ float sEm[EDG];
    __shared__ float sG [EDG];        // att gate, later coord weight

    const int tid = threadIdx.x;
    const int b = blockIdx.z, ti = blockIdx.y, tj = blockIdx.x;
    const int i0 = ti * TI, j0 = tj * TJ;

    // phase 0: stage d, edge mask, wd, We2
    if (tid < EDG) {
        int li = tid & 15, lj = tid >> 4;
        size_t eidx = ((size_t)b * Nk + (i0 + li)) * Nk + (j0 + lj);
        sD[tid]  = d2[eidx];
        sEm[tid] = emask[eidx];
        sVec[tid] = wd[tid];
    }
    for (int idx = tid; idx < Hk * Hk; idx += 256)
        sW[idx >> 7][idx & 127] = We2[idx];
    __syncthreads();

    // phase 1: m1 = silu(aT[i] + bT[j] + d*wd)
    for (int idx = tid; idx < EDG * Hk; idx += 256) {
        int e = idx >> 7, f = idx & 127;
        int li = e & 15, lj = e >> 4;
        float s = projA[((size_t)b * Nk + (i0 + li)) * Hk + f]
                + projB[((size_t)b * Nk + (j0 + lj)) * Hk + f]
                + sD[e] * sVec[f];
        sM1[e][f] = fast_silu(s);
    }
    __syncthreads();

    // phase 2: m2 = silu(m1 @ We2 + be2)    [WMMA]
    tile_mm_silu(&sM1[0][0], &sW[0][0], be2, &sM2[0][0], tid);
    __syncthreads();

    // phase 3: stage Wa and Wc1 (We2 no longer needed)
    if (tid < Hk) sVec[tid] = Wa[tid];
    for (int idx = tid; idx < Hk * Hk; idx += 256)
        sW[idx >> 7][idx & 127] = Wc1[idx];
    __syncthreads();

    if (tid < EDG) {                       // att = sigmoid(m2.Wa + ba) * emask
        float s = ba[0];
        for (int f = 0; f < Hk; ++f) s += sM2[tid][f] * sVec[f];
        sG[tid] = fast_sigmoid(s) * sEm[tid];
    }
    __syncthreads();

    // phase 4: mg = m2 * gate; stage Wc2
    for (int idx = tid; idx < EDG * Hk; idx += 256) {
        int e = idx >> 7;
        sM2[e][idx & 127] *= sG[e];
    }
    if (tid < Hk) sVec[tid] = Wc2[tid];
    __syncthreads();

    // phase 5: partial aggregation over this block's j-chunk (deterministic)
    {
        size_t base = (((size_t)b * NTI + ti) * NTJ + tj) * (size_t)(TI * Hk);
        for (int idx = tid; idx < TI * Hk; idx += 256) {
            int li = idx >> 7, f = idx & 127;
            float s = 0.0f;
            #pragma unroll
            for (int lj = 0; lj < TJ; ++lj) s += sM2[lj * 16 + li][f];
            aggp[base + idx] = s;
        }
    }
    __syncthreads();

    // phase 6: c = silu(mg @ Wc1 + bc1)    [WMMA]
    tile_mm_silu(&sM2[0][0], &sW[0][0], bc1, &sM1[0][0], tid);
    __syncthreads();

    if (tid < EDG) {                       // w = tanh(c.Wc2)*CR * emask
        float s = 0.0f;
        for (int f = 0; f < Hk; ++f) s += sM1[tid][f] * sVec[f];
        sG[tid] = tanhf(s) * CRANGE * sEm[tid];
    }
    __syncthreads();

    // phase 7: coordinate partials: sum_j cd_norm * w
    if (tid < TI * 3) {
        int li = tid / 3, dm = tid % 3;
        const float* xi = xcur + ((size_t)b * Nk + (i0 + li)) * 3;
        float x0 = xi[0], x1 = xi[1], x2 = xi[2];
        float acc = 0.0f;
        for (int lj = 0; lj < TJ; ++lj) {
            const float* xj = xcur + ((size_t)b * Nk + (j0 + lj)) * 3;
            float c0 = x0 - xj[0], c1 = x1 - xj[1], c2 = x2 - xj[2];
            float inv = __builtin_amdgcn_rcpf(sqrtf(c0*c0 + c1*c1 + c2*c2) + 1.0f);
            float cd = (dm == 0 ? c0 : (dm == 1 ? c1 : c2)) * inv;
            acc += cd * sG[lj * 16 + li];
        }
        xp[(((size_t)b * NTI + ti) * NTJ + tj) * (size_t)(TI * 3) + tid] = acc;
    }
}

// ---------------- node update (reduce + MLP + residual + coords) ----------------
__global__ __launch_bounds__(256) void node_kernel(
    float* __restrict__ hcur, float* __restrict__ xcur,
    const float* __restrict__ aggp, const float* __restrict__ xp,
    const float* __restrict__ nmask,
    const float* __restrict__ Wn1, const float* __restrict__ bn1,
    const float* __restrict__ Wn2, const float* __restrict__ bn2)
{
    __shared__ float sIn[TI][2 * Hk];
    __shared__ float sH1[TI][Hk];
    const int tid = threadIdx.x;
    const int b = blockIdx.y, ti = blockIdx.x;

    for (int idx = tid; idx < TI * Hk; idx += 256) {
        int li = idx >> 7, f = idx & 127;
        int i = ti * TI + li;
        sIn[li][f] = hcur[((size_t)b * Nk + i) * Hk + f];
        float s = 0.0f;
        size_t base = (((size_t)b * NTI + ti) * NTJ) * (size_t)(TI * Hk) + (size_t)li * Hk + f;
        for (int tjc = 0; tjc < NTJ; ++tjc) s += aggp[base + (size_t)tjc * TI * Hk];
        sIn[li][Hk + f] = s;
    }
    __syncthreads();

    for (int idx = tid; idx < TI * Hk; idx += 256) {
        int li = idx >> 7, o = idx & 127;
        float s = bn1[o];
        for (int k = 0; k < 2 * Hk; ++k) s += sIn[li][k] * Wn1[(size_t)k * Hk + o];
        sH1[li][o] = fast_silu(s);
    }
    __syncthreads();

    for (int idx = tid; idx < TI * Hk; idx += 256) {
        int li = idx >> 7, o = idx & 127;
        int i = ti * TI + li;
        float s = bn2[o];
        for (int k = 0; k < Hk; ++k) s += sH1[li][k] * Wn2[(size_t)k * Hk + o];
        size_t hidx = ((size_t)b * Nk + i) * Hk + o;
        hcur[hidx] = (hcur[hidx] + s) * nmask[(size_t)b * Nk + i];
    }

    if (tid < TI * 3) {
        int li = tid / 3, dm = tid % 3;
        int i = ti * TI + li;
        float s = xcur[((size_t)b * Nk + i) * 3 + dm];
        size_t base = (((size_t)b * NTI + ti) * NTJ) * (size_t)(TI * 3) + tid;
        for (int tjc = 0; tjc < NTJ; ++tjc) s += xp[base + (size_t)tjc * TI * 3];
        xcur[((size_t)b * Nk + i) * 3 + dm] = s;
    }
}

// ---------------- output projection + softmax ----------------
__global__ __launch_bounds__(32) void final_kernel(
    const float* __restrict__ hcur, const float* __restrict__ xcur,
    const float* __restrict__ nmask, const float* __restrict__ W_out,
    const float* __restrict__ b_out, float* __restrict__ out)
{
    __shared__ float sl[Fk];
    const int n = blockIdx.x;              // 0..B*N-1
    const int tid = threadIdx.x;
    const float nm = nmask[n];
    if (tid < Fk) {
        float s = b_out[tid];
        const float* hr = hcur + (size_t)n * Hk;
        for (int k = 0; k < Hk; ++k) s += hr[k] * W_out[(size_t)k * Fk + tid];
        sl[tid] = s * nm;
    }
    __syncthreads();
    if (tid < Fk) {
        float mx = sl[0];
        #pragma unroll
        for (int k = 1; k < Fk; ++k) mx = fmaxf(mx, sl[k]);
        float den = 0.0f;
        #pragma unroll
        for (int k = 0; k < Fk; ++k) den += __expf(sl[k] - mx);
        out[(size_t)n * Fk + tid] = __expf(sl[tid] - mx) / den;
    }
    if (tid >= 16 && tid < 19) {
        int dm = tid - 16;
        out[(size_t)Bk * Nk * Fk + (size_t)n * 3 + dm] = xcur[(size_t)n * 3 + dm] * nm;
    }
}

// ---------------- host ----------------
extern "C" void kernel_launch(void* const* d_in, const int* in_sizes, int n_in,
                              void* d_out, int out_size, void* d_ws, size_t ws_size,
                              hipStream_t stream)
{
    (void)out_size; (void)ws_size;
    const float* h     = (const float*)d_in[0];
    const float* x     = (const float*)d_in[1];
    // d_in[2]=hT, d_in[3]=xT: unused by reference
    const float* t     = (const float*)d_in[4];
    const float* nmask = (const float*)d_in[5];
    const float* emask = (const float*)d_in[6];
    const float* W_in  = (const float*)d_in[7];
    const float* b_in  = (const float*)d_in[8];
    const float* W_out = (const float*)d_in[9];
    const float* b_out = (const float*)d_in[10];

    // Resolve layer_params: P = {We1,be1,We2,be2,Wa,ba,Wn1,bn1,Wn2,bn2,Wc1,bc1,Wc2}
    const float* P[13];
    if (n_in >= 24) {
        if (in_sizes[11] == Lk * 257 * Hk) {           // insertion order
            for (int k = 0; k < 13; ++k) P[k] = (const float*)d_in[11 + k];
        } else {                                        // jax-sorted key order
            const float* Q[13];
            for (int k = 0; k < 13; ++k) Q[k] = (const float*)d_in[11 + k];
            P[0]=Q[3];  P[1]=Q[9];  P[2]=Q[4];  P[3]=Q[10]; P[4]=Q[0];  P[5]=Q[7];
            P[6]=Q[5];  P[7]=Q[11]; P[8]=Q[6];  P[9]=Q[12]; P[10]=Q[1]; P[11]=Q[8];
            P[12]=Q[2];
        }
    } else {                                            // single flat blob
        const float* base = (const float*)d_in[11];
        const size_t sz[13] = {131584,512,65536,512,512,4,131072,512,65536,512,65536,512,512};
        size_t off = 0;
        for (int k = 0; k < 13; ++k) { P[k] = base + off; off += sz[k]; }
    }

    // Workspace layout (floats)
    float* ws    = (float*)d_ws;
    float* hcur  = ws;                         // 81920
    float* xcur  = hcur + 81920;               // 1920
    float* d2    = xcur + 1920;                // 102400
    float* projA = d2   + 102400;              // 81920
    float* projB = projA + 81920;              // 81920
    float* aggp  = projB + 81920;              // 1638400
    float* xp    = aggp  + 1638400;            // 38400

    embed_kernel<<<320, 256, 0, stream>>>(h, t, nmask, W_in, b_in, hcur);
    dist_kernel<<<400, 256, 0, stream>>>(x, d2, xcur);

    for (int l = 0; l < Lk; ++l) {
        const float* We1 = P[0]  + (size_t)l * 257 * Hk;
        const float* be1 = P[1]  + (size_t)l * Hk;
        const float* We2 = P[2]  + (size_t)l * Hk * Hk;
        const float* be2 = P[3]  + (size_t)l * Hk;
        const float* Wa  = P[4]  + (size_t)l * Hk;
        const float* ba  = P[5]  + (size_t)l * 1;
        const float* Wn1 = P[6]  + (size_t)l * 2 * Hk * Hk;
        const float* bn1 = P[7]  + (size_t)l * Hk;
        const float* Wn2 = P[8]  + (size_t)l * Hk * Hk;
        const float* bn2 = P[9]  + (size_t)l * Hk;
        const float* Wc1 = P[10] + (size_t)l * Hk * Hk;
        const float* bc1 = P[11] + (size_t)l * Hk;
        const float* Wc2 = P[12] + (size_t)l * Hk;
        const float* wd  = We1 + (size_t)256 * Hk;     // last row of We1 (distance term)

        proj_kernel<<<640, 256, 0, stream>>>(hcur, We1, be1, projA, projB);
        edge_kernel<<<dim3(NTJ, NTI, Bk), 256, 0, stream>>>(
            projA, projB, d2, xcur, emask, wd,
            We2, be2, Wa, ba, Wc1, bc1, Wc2, aggp, xp);
        node_kernel<<<dim3(NTI, Bk), 256, 0, stream>>>(
            hcur, xcur, aggp, xp, nmask, Wn1, bn1, Wn2, bn2);
    }

    final_kernel<<<Bk * Nk, 32, 0, stream>>>(hcur, xcur, nmask, W_out, b_out, (float*)d_out);
}